// GraphEncoder_30219389895226
// MI455X (gfx1250) — compile-verified
//
#include <hip/hip_runtime.h>

typedef __attribute__((ext_vector_type(2))) float v2f;
typedef __attribute__((ext_vector_type(8))) float v8f;

#define DIN  128
#define DH   256
#define DOUT 384

// ---------------- utility kernels ----------------

__global__ __launch_bounds__(256) void zero_kernel(float4* __restrict__ p, int n4) {
  int i = blockIdx.x * blockDim.x + threadIdx.x;
  if (i < n4) p[i] = make_float4(0.f, 0.f, 0.f, 0.f);
}

__global__ __launch_bounds__(256) void count_kernel(const int* __restrict__ dst,
                                                    int* __restrict__ cnt, int nedges) {
  int e = blockIdx.x * blockDim.x + threadIdx.x;
  if (e < nedges) atomicAdd(&cnt[dst[e]], 1);
}

// ---- counting-sort scan (N <= 65536: 256-wide blocks, <=256 partials) ----

__global__ __launch_bounds__(256) void scan_blocks_kernel(const int* __restrict__ cnt,
                                                          int* __restrict__ excl,
                                                          int* __restrict__ partials, int n) {
  __shared__ int sh[256];
  const int t = threadIdx.x;
  const int i = blockIdx.x * 256 + t;
  int v = (i < n) ? cnt[i] : 0;
  sh[t] = v;
  __syncthreads();
  for (int o = 1; o < 256; o <<= 1) {
    int x = (t >= o) ? sh[t - o] : 0;
    __syncthreads();
    sh[t] += x;
    __syncthreads();
  }
  if (i < n) excl[i] = sh[t] - v;              // exclusive scan within block
  if (t == 255) partials[blockIdx.x] = sh[255];
}

__global__ __launch_bounds__(256) void scan_partials_kernel(int* __restrict__ partials, int nb) {
  __shared__ int sh[256];
  const int t = threadIdx.x;
  int v = (t < nb) ? partials[t] : 0;
  sh[t] = v;
  __syncthreads();
  for (int o = 1; o < 256; o <<= 1) {
    int x = (t >= o) ? sh[t - o] : 0;
    __syncthreads();
    sh[t] += x;
    __syncthreads();
  }
  if (t < nb) partials[t] = sh[t] - v;         // exclusive block offsets
}

__global__ __launch_bounds__(256) void scan_finish_kernel(int* __restrict__ rowptr,
                                                          const int* __restrict__ partials,
                                                          const int* __restrict__ cnt,
                                                          int* __restrict__ cursor,
                                                          float* __restrict__ invdeg, int n) {
  int i = blockIdx.x * 256 + threadIdx.x;
  if (i >= n) return;
  int r = rowptr[i] + partials[i >> 8];
  rowptr[i] = r;
  cursor[i] = r;
  invdeg[i] = 1.0f / fmaxf((float)cnt[i], 1.0f);
  if (i == n - 1) rowptr[n] = r + cnt[i];
}

__global__ __launch_bounds__(256) void fill_adj_kernel(const int* __restrict__ src,
                                                       const int* __restrict__ dst,
                                                       int* __restrict__ cursor,
                                                       int* __restrict__ adj, int nedges) {
  int e = blockIdx.x * blockDim.x + threadIdx.x;
  if (e >= nedges) return;
  int pos = atomicAdd(&cursor[dst[e]], 1);
  adj[pos] = src[e];
}

// ---------------- CSR gather: mean[i] = invdeg[i] * sum_{s in adj[i]} h[s] ----------------
// One wave per (node, 64-feature chunk): each neighbor row chunk is a 256B
// coalesced read; accumulate in registers -> zero float atomics, mean fused.
template<int D>
__global__ __launch_bounds__(256) void gather_mean_kernel(const float* __restrict__ h,
                                                          const int* __restrict__ rowptr,
                                                          const int* __restrict__ adj,
                                                          const float* __restrict__ invdeg,
                                                          float* __restrict__ mean, int nnodes) {
  const int lane = threadIdx.x & 31;
  const int wave = threadIdx.x >> 5;
  const int node = blockIdx.x * 8 + wave;
  if (node >= nnodes) return;
  const int cbase = blockIdx.y * 64 + lane * 2;
  const int beg = rowptr[node];
  const int end = rowptr[node + 1];
  float2 acc = make_float2(0.f, 0.f);
  for (int j = beg; j < end; ++j) {
    const int s = adj[j];
    const float2 v = *(const float2*)(h + (size_t)s * D + cbase);
    acc.x += v.x;
    acc.y += v.y;
  }
  const float sc = invdeg[node];
  acc.x *= sc;
  acc.y *= sc;
  *(float2*)(mean + (size_t)node * D + cbase) = acc;
}

// ---------------- weight permute ----------------
// W[k][col] -> Wt laid out so lane (with stream s = lane>>4) reads its
// fragment pair (W[kk][col], W[kk+1][col]) as ONE contiguous b64, streaming
// sequentially over the K loop:  v2-index = (col*2 + stream)*(K/4) + j
// where stream=(k>>1)&1, j=k>>2, pos=k&1.
template<int K, int NOUT>
__global__ __launch_bounds__(256) void permute_w_kernel(const float* __restrict__ W,
                                                        float* __restrict__ Wt) {
  int i = blockIdx.x * blockDim.x + threadIdx.x;
  if (i >= K * NOUT) return;
  int k = i / NOUT;
  int col = i - k * NOUT;
  int stream = (k >> 1) & 1;
  int j = k >> 2;
  int pos = k & 1;
  Wt[((size_t)(col * 2 + stream) * (K / 4) + j) * 2 + pos] = W[i];
}

// ---------------- fused SAGE GEMM: out = mean@Wl + hin@Wr + b ----------------
// wave32 WMMA f32 16x16x4; each wave owns a 16x64 output tile. 5 b64 loads
// per 4 WMMA (1 A + 4 streaming B), bias+ReLU fused into the store.
template<int K, int NOUT, bool RELU>
__global__ __launch_bounds__(256) void sage_gemm_kernel(const float* __restrict__ mean,
                                                        const float* __restrict__ hin,
                                                        const float* __restrict__ Wlp,
                                                        const float* __restrict__ Wrp,
                                                        const float* __restrict__ bias,
                                                        float* __restrict__ out, int mtiles) {
  const int lane  = threadIdx.x & 31;
  const int wave  = threadIdx.x >> 5;
  const int mtile = blockIdx.x * 8 + wave;
  if (mtile >= mtiles) return;           // whole-wave exit: EXEC stays all-ones
  const int nbase = blockIdx.y * 64;
  const int l15   = lane & 15;
  const int row   = mtile * 16 + l15;
  const int s     = lane >> 4;           // B stream / A half
  const int khalf = s << 1;              // lanes 16-31 hold K=k0+2,k0+3

  v8f acc0 = {}, acc1 = {}, acc2 = {}, acc3 = {};

  for (int pass = 0; pass < 2; ++pass) {
    const float* __restrict__ Ar = (pass == 0 ? mean : hin) + (size_t)row * K;
    const float* __restrict__ Wp = (pass == 0 ? Wlp : Wrp);
    const v2f* bp0 = (const v2f*)Wp + (size_t)((nbase + l15 +  0) * 2 + s) * (K / 4);
    const v2f* bp1 = (const v2f*)Wp + (size_t)((nbase + l15 + 16) * 2 + s) * (K / 4);
    const v2f* bp2 = (const v2f*)Wp + (size_t)((nbase + l15 + 32) * 2 + s) * (K / 4);
    const v2f* bp3 = (const v2f*)Wp + (size_t)((nbase + l15 + 48) * 2 + s) * (K / 4);
#pragma unroll 4
    for (int j = 0; j < K / 4; ++j) {
      v2f a = *(const v2f*)(Ar + 4 * j + khalf);
      acc0 = __builtin_amdgcn_wmma_f32_16x16x4_f32(false, a, false, bp0[j], (short)0, acc0, false, false);
      acc1 = __builtin_amdgcn_wmma_f32_16x16x4_f32(false, a, false, bp1[j], (short)0, acc1, false, false);
      acc2 = __builtin_amdgcn_wmma_f32_16x16x4_f32(false, a, false, bp2[j], (short)0, acc2, false, false);
      acc3 = __builtin_amdgcn_wmma_f32_16x16x4_f32(false, a, false, bp3[j], (short)0, acc3, false, false);
    }
  }

  // C/D layout: VGPR v -> row = v (+8 for lanes 16-31), col = lane&15
  const int col = nbase + l15;
  const float bj0 = bias[col];
  const float bj1 = bias[col + 16];
  const float bj2 = bias[col + 32];
  const float bj3 = bias[col + 48];
  const int r0 = mtile * 16 + (s << 3);
#pragma unroll
  for (int v = 0; v < 8; ++v) {
    float* o = out + (size_t)(r0 + v) * NOUT + col;
    float t0 = acc0[v] + bj0;
    float t1 = acc1[v] + bj1;
    float t2 = acc2[v] + bj2;
    float t3 = acc3[v] + bj3;
    if (RELU) {
      t0 = fmaxf(t0, 0.f); t1 = fmaxf(t1, 0.f);
      t2 = fmaxf(t2, 0.f); t3 = fmaxf(t3, 0.f);
    }
    o[0] = t0; o[16] = t1; o[32] = t2; o[48] = t3;
  }
}

// ---------------- host launch ----------------

extern "C" void kernel_launch(void* const* d_in, const int* in_sizes, int n_in,
                              void* d_out, int out_size, void* d_ws, size_t ws_size,
                              hipStream_t stream) {
  const float* x   = (const float*)d_in[0];
  const int*   ei  = (const int*)d_in[1];
  const float* W1l = (const float*)d_in[2];
  const float* W1r = (const float*)d_in[3];
  const float* b1  = (const float*)d_in[4];
  const float* W2l = (const float*)d_in[5];
  const float* W2r = (const float*)d_in[6];
  const float* b2  = (const float*)d_in[7];
  const float* W3l = (const float*)d_in[8];
  const float* W3r = (const float*)d_in[9];
  const float* b3  = (const float*)d_in[10];
  float* out = (float*)d_out;

  const int N = in_sizes[0] / DIN;   // 50000 (divisible by 16)
  const int E = in_sizes[1] / 2;     // 800000
  const int* src = ei;
  const int* dst = ei + E;

  // workspace carve-out (~158 MB)
  char* ws = (char*)d_ws;
  auto al = [](size_t v) { return (v + 255) & ~(size_t)255; };
  size_t off = 0;
  int*   cnt      = (int*)(ws + off);   off += al((size_t)N * 4);
  int*   rowptr   = (int*)(ws + off);   off += al((size_t)(N + 1) * 4);
  int*   cursor   = (int*)(ws + off);   off += al((size_t)N * 4);
  int*   partials = (int*)(ws + off);   off += al((size_t)256 * 4);
  float* invdeg   = (float*)(ws + off); off += al((size_t)N * 4);
  int*   adj      = (int*)(ws + off);   off += al((size_t)E * 4);
  float* mean     = (float*)(ws + off); off += al((size_t)N * DH * 4);
  float* h1       = (float*)(ws + off); off += al((size_t)N * DH * 4);
  float* h2       = (float*)(ws + off); off += al((size_t)N * DH * 4);
  float* Wt1l     = (float*)(ws + off); off += al((size_t)DIN * DH * 4);
  float* Wt1r     = (float*)(ws + off); off += al((size_t)DIN * DH * 4);
  float* Wt2l     = (float*)(ws + off); off += al((size_t)DH * DH * 4);
  float* Wt2r     = (float*)(ws + off); off += al((size_t)DH * DH * 4);
  float* Wt3l     = (float*)(ws + off); off += al((size_t)DH * DOUT * 4);
  float* Wt3r     = (float*)(ws + off);

  const int TB = 256;
  const int NB = (N + 255) / 256;
  const int mtiles = N / 16;
  const int gm = (mtiles + 7) / 8;
  const int gn = (N + 7) / 8;

  // ---- CSR-by-dst build (counting sort) + inv_deg ----
  zero_kernel<<<(N / 4 + TB - 1) / TB, TB, 0, stream>>>((float4*)cnt, N / 4);
  count_kernel<<<(E + TB - 1) / TB, TB, 0, stream>>>(dst, cnt, E);
  scan_blocks_kernel<<<NB, TB, 0, stream>>>(cnt, rowptr, partials, N);
  scan_partials_kernel<<<1, TB, 0, stream>>>(partials, NB);
  scan_finish_kernel<<<NB, TB, 0, stream>>>(rowptr, partials, cnt, cursor, invdeg, N);
  fill_adj_kernel<<<(E + TB - 1) / TB, TB, 0, stream>>>(src, dst, cursor, adj, E);

  // ---- weight permutes (tiny, one-shot) ----
  permute_w_kernel<DIN, DH ><<<(DIN * DH  + TB - 1) / TB, TB, 0, stream>>>(W1l, Wt1l);
  permute_w_kernel<DIN, DH ><<<(DIN * DH  + TB - 1) / TB, TB, 0, stream>>>(W1r, Wt1r);
  permute_w_kernel<DH,  DH ><<<(DH  * DH  + TB - 1) / TB, TB, 0, stream>>>(W2l, Wt2l);
  permute_w_kernel<DH,  DH ><<<(DH  * DH  + TB - 1) / TB, TB, 0, stream>>>(W2r, Wt2r);
  permute_w_kernel<DH, DOUT><<<(DH * DOUT + TB - 1) / TB, TB, 0, stream>>>(W3l, Wt3l);
  permute_w_kernel<DH, DOUT><<<(DH * DOUT + TB - 1) / TB, TB, 0, stream>>>(W3r, Wt3r);

  // ---- Layer 1: mean over x (K=DIN) -> h1 = relu(mean@W1l + x@W1r + b1) ----
  {
    dim3 ga(gn, DIN / 64);
    gather_mean_kernel<DIN><<<ga, TB, 0, stream>>>(x, rowptr, adj, invdeg, mean, N);
    dim3 g(gm, DH / 64);
    sage_gemm_kernel<DIN, DH, true><<<g, TB, 0, stream>>>(mean, x, Wt1l, Wt1r, b1, h1, mtiles);
  }
  // ---- Layer 2 ----
  {
    dim3 ga(gn, DH / 64);
    gather_mean_kernel<DH><<<ga, TB, 0, stream>>>(h1, rowptr, adj, invdeg, mean, N);
    dim3 g(gm, DH / 64);
    sage_gemm_kernel<DH, DH, true><<<g, TB, 0, stream>>>(mean, h1, Wt2l, Wt2r, b2, h2, mtiles);
  }
  // ---- Layer 3 -> out (no relu) ----
  {
    dim3 ga(gn, DH / 64);
    gather_mean_kernel<DH><<<ga, TB, 0, stream>>>(h2, rowptr, adj, invdeg, mean, N);
    dim3 g(gm, DOUT / 64);
    sage_gemm_kernel<DH, DOUT, false><<<g, TB, 0, stream>>>(mean, h2, Wt3l, Wt3r, b3, out, mtiles);
  }
  (void)n_in; (void)out_size; (void)ws_size;
}